// RolloutEncoder_8237747274253
// MI455X (gfx1250) — compile-verified
//
#include <hip/hip_runtime.h>
#include <hip/hip_bf16.h>

// ---------------------------------------------------------------------------
// RolloutEncoder GRU for MI455X (gfx1250, wave32, WMMA + async LDS DMA)
//
// Phase 1: gi = x @ w_ih^T + b_ih   (M=8192, N=3072, K=2048(+1 reward col))
//          fp32 WMMA (v_wmma_f32_16x16x4_f32), double-buffered LDS tiles
//          filled with GLOBAL_LOAD_ASYNC_TO_LDS (ASYNCcnt), compute bound.
// Phase 2: reverse GRU scan, 8192 sequential steps. 64 persistent WGs,
//          w_hh stays L2-resident (12.6 MB << 192 MB L2), one grid barrier
//          per step, h double-buffered in global scratch.
// ---------------------------------------------------------------------------

typedef float v2f __attribute__((ext_vector_type(2)));
typedef float v8f __attribute__((ext_vector_type(8)));
typedef int   v4i __attribute__((ext_vector_type(4)));

#define T_STEPS 8192
#define HID     1024
#define NGATES  3072      // 3*HID
#define KF      2048      // WMMA K extent (feature part)
#define INSZ    2049      // w_ih row length (features + reward col)

// Async global->LDS DMA: builtin signature is (AS1 typed ptr, AS3 typed ptr,
// imm offset, imm cpol) — pointee types are int4 (b128) / int (b32).
#define ASYNC_CP_B128(g, l)                                        \
    __builtin_amdgcn_global_load_async_to_lds_b128(                \
        (__attribute__((address_space(1))) v4i*)(g),               \
        (__attribute__((address_space(3))) v4i*)(l), 0, 0)
#define ASYNC_CP_B32(g, l)                                         \
    __builtin_amdgcn_global_load_async_to_lds_b32(                 \
        (__attribute__((address_space(1))) int*)(g),               \
        (__attribute__((address_space(3))) int*)(l), 0, 0)
#define WAIT_ASYNC() asm volatile("s_wait_asynccnt 0x0" ::: "memory")

// -------------------- phase 1: WMMA GEMM tiling ----------------------------
#define BM 64
#define BN 64
#define BK 32
#define LDP (BK + 4)      // padded row stride: 36 floats -> conflict-free,
                          // 144 B rows keep 16 B alignment for b128 DMA

// Issue the async global->LDS DMAs for one (A,B) tile pair.
//   A: 64x32 floats, 2 x b128 per thread (feat rows are 16B-aligned).
//   B: stored as Bdst[n][k] = w_ih[n0+n][k0+k]; odd row stride 2049 forces
//      b32 granularity (8 per thread), coalesced in k across each wave.
__device__ __forceinline__ void load_tiles_async(
    const float* __restrict__ featTile,   // &feat[m0*KF + k0], row stride KF
    const float* __restrict__ wihTile,    // &w_ih[n0*INSZ + k0], row stride INSZ
    float (*Adst)[LDP], float (*Bdst)[LDP], int tid)
{
#pragma unroll
    for (int i = 0; i < 2; ++i) {
        int idx = tid + i * 256;          // 0..511 float4 slots
        int m   = idx >> 3;               // 8 float4 per 32-float row
        int kq  = (idx & 7) << 2;
        ASYNC_CP_B128(featTile + (size_t)m * KF + kq, &Adst[m][kq]);
    }
#pragma unroll
    for (int i = 0; i < 8; ++i) {
        int idx = tid + i * 256;          // 0..2047
        int j   = idx >> 5;               // output column n (0..63)
        int k   = idx & 31;               // k within tile (coalesced per wave)
        ASYNC_CP_B32(wihTile + (size_t)j * INSZ + k, &Bdst[j][k]);
    }
}

__global__ __launch_bounds__(256)
void gi_gemm_wmma(const float* __restrict__ feat,   // [T, 2048]
                  const float* __restrict__ rew,    // [T]
                  const float* __restrict__ w_ih,   // [3072, 2049]
                  const float* __restrict__ b_ih,   // [3072]
                  float* __restrict__ gi)           // [T, 3072]
{
    __shared__ float Asl[2][BM][LDP];     // double-buffered: 2 x 9 KB
    __shared__ float Bsl[2][BN][LDP];     //                  2 x 9 KB

    const int tid  = threadIdx.x;
    const int n0   = blockIdx.x * BN;     // gate-row (N) tile
    const int m0   = blockIdx.y * BM;     // timestep (M) tile

    const int wave = tid >> 5;            // 0..7
    const int lane = tid & 31;
    const int half = lane >> 4;           // 0: lanes 0-15, 1: lanes 16-31
    const int l16  = lane & 15;

    const int waveM = wave >> 1;          // 0..3 -> M offset 16*waveM
    const int waveN = wave & 1;           // 0..1 -> N offsets 32*waveN + {0,16}

    const float* featRow = feat + (size_t)m0 * KF;
    const float* wihRow  = w_ih + (size_t)n0 * INSZ;

    v8f c0 = {};                          // two 16x16 fp32 accumulators / wave
    v8f c1 = {};

    // Prologue: DMA the first tile pair, then wait + barrier.
    load_tiles_async(featRow, wihRow, Asl[0], Bsl[0], tid);
    WAIT_ASYNC();
    __syncthreads();

    for (int k0 = 0; k0 < KF; k0 += BK) {
        const int cur = (k0 >> 5) & 1;

        // Kick off next tile's DMA while this tile computes.
        if (k0 + BK < KF)
            load_tiles_async(featRow + (k0 + BK), wihRow + (k0 + BK),
                             Asl[cur ^ 1], Bsl[cur ^ 1], tid);

        // 8 x (K=4) WMMA steps. f32 16x16x4 lane layout:
        //   A: lane m=l16 holds K pair starting at 2*half  -> one b64 read
        //   B: vgpr pair holds rows {2*half, 2*half+1} at column n=l16
        //      -> Bsl[n][k2..k2+1], one b64 read, no repacking movs.
#pragma unroll
        for (int kk = 0; kk < BK; kk += 4) {
            const int k2 = kk + 2 * half;
            v2f a  = *(const v2f*)(&Asl[cur][16 * waveM + l16][k2]);
            v2f b0 = *(const v2f*)(&Bsl[cur][32 * waveN + l16][k2]);
            v2f b1 = *(const v2f*)(&Bsl[cur][32 * waveN + 16 + l16][k2]);
            c0 = __builtin_amdgcn_wmma_f32_16x16x4_f32(
                    false, a, false, b0, (short)0, c0, false, false);
            c1 = __builtin_amdgcn_wmma_f32_16x16x4_f32(
                    false, a, false, b1, (short)0, c1, false, false);
        }

        // Next buffer must be fully DMA'd (all waves) before it is read.
        WAIT_ASYNC();
        __syncthreads();
    }

    // Epilogue: + bias + reward * w_ih[:,2048]  (folds the K=2049th column)
    // C/D layout: vgpr r -> row = r + 8*half, col = l16 (per 16x16 tile).
    const int mrow = m0 + 16 * waveM + 8 * half;
    {
        int col = n0 + 32 * waveN + l16;
        float wlast = w_ih[(size_t)col * INSZ + KF];
        float bias  = b_ih[col];
#pragma unroll
        for (int r = 0; r < 8; ++r) {
            int row = mrow + r;
            gi[(size_t)row * NGATES + col] = c0[r] + bias + rew[row] * wlast;
        }
    }
    {
        int col = n0 + 32 * waveN + 16 + l16;
        float wlast = w_ih[(size_t)col * INSZ + KF];
        float bias  = b_ih[col];
#pragma unroll
        for (int r = 0; r < 8; ++r) {
            int row = mrow + r;
            gi[(size_t)row * NGATES + col] = c1[r] + bias + rew[row] * wlast;
        }
    }
}

// -------------------- phase 2: persistent GRU scan -------------------------
#define NWG   64          // persistent workgroups (must be co-resident)
#define SLICE 16          // hidden units owned per WG (1024/64)

__device__ __forceinline__ float sigm(float x) {
    return 1.0f / (1.0f + __expf(-x));
}

__global__ __launch_bounds__(256)
void gru_scan(const float* __restrict__ gi,     // [T, 3072]
              const float* __restrict__ w_hh,   // [3072, 1024] (L2-resident)
              const float* __restrict__ b_hh,   // [3072]
              float* __restrict__ hbuf,         // [2][1024] double buffer
              int*   __restrict__ ctr,          // grid barrier counter
              float* __restrict__ out)          // [1024]
{
    __shared__ float hs[HID];
    __shared__ float dots[3][SLICE];

    const int tid   = threadIdx.x;
    const int wg    = blockIdx.x;
    const int hbase = wg * SLICE;
    const int grp   = tid >> 4;        // 0..15: owned hidden index
    const int l16   = tid & 15;        // lane within 16-lane dot group

    const float* wr = w_hh + (size_t)(hbase + grp) * HID;
    const float* wz = wr + (size_t)HID * HID;
    const float* wn = wz + (size_t)HID * HID;

    for (int iter = 0; iter < T_STEPS; ++iter) {
        const int t = T_STEPS - 1 - iter;            // reverse scan
        const float* hin  = hbuf + (iter & 1) * HID;
        float*       hout = hbuf + ((iter & 1) ^ 1) * HID;

        // h_t -> LDS: one async b128 DMA per lane (4 KB total, no VGPRs).
        ASYNC_CP_B128(hin + 4 * tid, &hs[4 * tid]);

        // Prefetch next step's gi row (backward stride; HW won't prefetch it)
        if (t > 0 && tid == 0)
            __builtin_prefetch(gi + (size_t)(t - 1) * NGATES + hbase, 0, 0);

        WAIT_ASYNC();
        __syncthreads();

        // Three 1024-long dots per owned unit; 16 lanes each, coalesced k.
        float accR = 0.f, accZ = 0.f, accN = 0.f;
#pragma unroll 4
        for (int k = l16; k < HID; k += 16) {
            float hv = hs[k];
            accR = fmaf(wr[k], hv, accR);
            accZ = fmaf(wz[k], hv, accZ);
            accN = fmaf(wn[k], hv, accN);
        }
#pragma unroll
        for (int m = 8; m >= 1; m >>= 1) {           // reduce within 16 lanes
            accR += __shfl_xor(accR, m, 32);
            accZ += __shfl_xor(accZ, m, 32);
            accN += __shfl_xor(accN, m, 32);
        }
        if (l16 == 0) { dots[0][grp] = accR; dots[1][grp] = accZ; dots[2][grp] = accN; }
        __syncthreads();

        // Gate math + state update on 16 lanes
        if (tid < SLICE) {
            const int    i  = tid;
            const size_t gb = (size_t)t * NGATES;
            float gr = gi[gb +            hbase + i];
            float gz = gi[gb +     HID  + hbase + i];
            float gn = gi[gb + 2 * HID  + hbase + i];
            float r  = sigm(gr + dots[0][i] + b_hh[          hbase + i]);
            float z  = sigm(gz + dots[1][i] + b_hh[    HID + hbase + i]);
            float n  = tanhf(gn + r * (dots[2][i] + b_hh[2 * HID + hbase + i]));
            float hp = hs[hbase + i];
            float hv = (1.f - z) * n + z * hp;
            hout[hbase + i] = hv;
            if (iter == T_STEPS - 1) out[hbase + i] = hv;   // final h -> d_out
        }

        // ---- device-wide barrier: release h writes, spin with s_sleep ----
        __syncthreads();
        if (tid == 0) {
            __threadfence();
            atomicAdd(ctr, 1);
            const int target = NWG * (iter + 1);
            while (__hip_atomic_load(ctr, __ATOMIC_ACQUIRE,
                                     __HIP_MEMORY_SCOPE_AGENT) < target) {
                __builtin_amdgcn_s_sleep(1);
            }
        }
        __syncthreads();
    }
}

// -------------------- scratch init (runs every launch) ---------------------
__global__ void scan_init(float* hbuf, int* ctr)
{
    int t = blockIdx.x * blockDim.x + threadIdx.x;
    if (t < HID) hbuf[t] = 0.0f;       // h_T = 0 (buffer 0 is read first)
    if (t == 0)  *ctr = 0;             // barrier counter
}

// ---------------------------------------------------------------------------
extern "C" void kernel_launch(void* const* d_in, const int* in_sizes, int n_in,
                              void* d_out, int out_size, void* d_ws, size_t ws_size,
                              hipStream_t stream)
{
    const float* feat = (const float*)d_in[0];   // [8192, 2048]
    const float* rew  = (const float*)d_in[1];   // [8192]
    const float* w_ih = (const float*)d_in[2];   // [3072, 2049]
    const float* w_hh = (const float*)d_in[3];   // [3072, 1024]
    const float* b_ih = (const float*)d_in[4];   // [3072]
    const float* b_hh = (const float*)d_in[5];   // [3072]
    float* out = (float*)d_out;                  // [1024]

    // Workspace layout: gi [T,3072] | h double buffer [2][1024] | counter
    float* gi   = (float*)d_ws;
    float* hbuf = gi + (size_t)T_STEPS * NGATES;
    int*   ctr  = (int*)(hbuf + 2 * HID);

    scan_init<<<4, 256, 0, stream>>>(hbuf, ctr);

    dim3 g1(NGATES / BN, T_STEPS / BM);          // 48 x 128 blocks
    gi_gemm_wmma<<<g1, 256, 0, stream>>>(feat, rew, w_ih, b_ih, gi);

    gru_scan<<<NWG, 256, 0, stream>>>(gi, w_hh, b_hh, hbuf, ctr, out);
}